// LMEncoder_43808666419692
// MI455X (gfx1250) — compile-verified
//
#include <hip/hip_runtime.h>

// Problem sizes (fixed by the reference)
#define BB 16
#define SS 512
#define HH 768
#define TROWS 20            // K window: 16 t-rows + span(3) + pad to multiple of 4
#define LDS_STRIDE 784      // floats per LDS row; rows 2 apart -> 32 banks apart (conflict-free B reads)

typedef __attribute__((ext_vector_type(2))) float v2f;
typedef __attribute__((ext_vector_type(4))) float v4f;
typedef __attribute__((ext_vector_type(8))) float v8f;

// ---------------------------------------------------------------------------
// Kernel 1: mask_len[b] = sum(masks[b, :])   (16 blocks x 256 threads)
// ---------------------------------------------------------------------------
__global__ __launch_bounds__(256) void masklen_kernel(const int* __restrict__ masks,
                                                      int* __restrict__ mlen) {
    __shared__ int red[256];
    const int b = blockIdx.x;
    const int t = threadIdx.x;
    red[t] = masks[b * SS + t] + masks[b * SS + 256 + t];
    __syncthreads();
    for (int off = 128; off > 0; off >>= 1) {
        if (t < off) red[t] += red[t + off];
        __syncthreads();
    }
    if (t == 0) mlen[b] = red[0];
}

// ---------------------------------------------------------------------------
// Kernel 2: banded span-sum pooling as a 16x16x20 WMMA chain per tile.
//   out[b,t,:] = 0.25 * sum_{j=t+1}^{t+span} sum_l hidden[l,b,j,:]  (valid t)
// Grid: (S/16 t-tiles, B).  Block: 256 threads = 8 waves.
//   - whole block stages Xsum (20 rows x 768, 4-layer sums) into LDS (float4 loads)
//   - each wave owns 6 of the 48 n-tiles; 5x v_wmma_f32_16x16x4_f32 per tile
// ---------------------------------------------------------------------------
__global__ __launch_bounds__(256) void span_pool_wmma(const float* __restrict__ hidden,
                                                      const int* __restrict__ spans,
                                                      const int* __restrict__ mlen,
                                                      float* __restrict__ out) {
    __shared__ float xsum[TROWS * LDS_STRIDE];   // 62,720 B

    const int tt  = blockIdx.x;          // t-tile index
    const int b   = blockIdx.y;          // batch
    const int t0  = tt * 16;
    const int tid = threadIdx.x;

    // ---- Stage: xsum[k][h] = sum over 4 layers of hidden[l,b,t0+1+k,h] ----
    const v4f* __restrict__ h4 = (const v4f*)hidden;
    for (int pos = tid; pos < TROWS * (HH / 4); pos += 256) {
        const int k  = pos / (HH / 4);
        const int c4 = pos % (HH / 4);
        const int j  = t0 + 1 + k;
        v4f acc = (v4f){0.f, 0.f, 0.f, 0.f};
        if (j < SS) {
#pragma unroll
            for (int l = 0; l < 4; ++l)
                acc += h4[((l * BB + b) * SS + j) * (HH / 4) + c4];
        }
        *(v4f*)&xsum[k * LDS_STRIDE + c4 * 4] = acc;   // zero rows beyond S (avoid 0*Inf NaN)
    }
    __syncthreads();

    const int lane = tid & 31;
    const int wave = tid >> 5;
    const int hi   = lane >> 4;          // which half-wave (selects K pair)
    const int nl   = lane & 15;          // A-row m / B,C,D column n

    // ---- Per-lane A fragments: A[m,k] = 0.25 iff m <= k < m+span (valid row) ----
    const int  t        = t0 + nl;
    const int  span     = spans[b * SS + t];
    const int  ml       = mlen[b];
    const bool rowValid = (t >= 1) && (t < ml - 1);
    v2f afrag[5];
#pragma unroll
    for (int kb = 0; kb < 5; ++kb) {
        const int k0 = kb * 4 + 2 * hi;
        const float ax = (rowValid && k0     >= nl && k0     < nl + span) ? 0.25f : 0.0f;
        const float ay = (rowValid && k0 + 1 >= nl && k0 + 1 < nl + span) ? 0.25f : 0.0f;
        afrag[kb] = (v2f){ax, ay};
    }

    // ---- WMMA over n-tiles: D(16x16) = A(16x20) x Xsum(20x16) ----
    for (int nt = wave; nt < HH / 16; nt += 8) {
        const int col = nt * 16 + nl;
        v8f c = {};
#pragma unroll
        for (int kb = 0; kb < 5; ++kb) {
            const int r = kb * 4 + 2 * hi;
            const v2f bfrag = (v2f){xsum[r * LDS_STRIDE + col],
                                    xsum[(r + 1) * LDS_STRIDE + col]};
            // (neg_a, A, neg_b, B, c_mod, C, reuse_a, reuse_b)
            c = __builtin_amdgcn_wmma_f32_16x16x4_f32(
                    false, afrag[kb], false, bfrag, (short)0, c, false, false);
        }
#pragma unroll
        for (int i = 0; i < 8; ++i) {
            const int row = t0 + i + 8 * hi;           // D layout: VGPR i -> M = i + 8*hi
            out[(b * SS + row) * HH + col] = c[i];
        }
    }
}

// ---------------------------------------------------------------------------
extern "C" void kernel_launch(void* const* d_in, const int* in_sizes, int n_in,
                              void* d_out, int out_size, void* d_ws, size_t ws_size,
                              hipStream_t stream) {
    const float* hidden = (const float*)d_in[0];   // [4,B,S,H] f32
    const int*   spans  = (const int*)d_in[1];     // [B,S] i32
    const int*   masks  = (const int*)d_in[2];     // [B,S] i32
    float*       out    = (float*)d_out;           // [B,S,H] f32
    int*         mlen   = (int*)d_ws;              // [B] scratch

    masklen_kernel<<<BB, 256, 0, stream>>>(masks, mlen);
    span_pool_wmma<<<dim3(SS / 16, BB), 256, 0, stream>>>(hidden, spans, mlen, out);
}